// RWKV_6_58918361366747
// MI455X (gfx1250) — compile-verified
//
#include <hip/hip_runtime.h>
#include <stdint.h>

// CDNA5 / gfx1250, wave32. FP32 WMMA: V_WMMA_F32_16X16X4_F32.
typedef float v2f __attribute__((ext_vector_type(2)));
typedef float v8f __attribute__((ext_vector_type(8)));

#define WMMA_F32(a, b, c) \
  __builtin_amdgcn_wmma_f32_16x16x4_f32(false, (a), false, (b), (short)0, (c), false, false)

#if __has_builtin(__builtin_amdgcn_global_load_async_to_lds_b128) && \
    __has_builtin(__builtin_amdgcn_s_wait_asynccnt)
#define ASYNC_LDS 1
#else
#define ASYNC_LDS 0
#endif

constexpr int B_ = 4, H_ = 32, T_ = 2048, K_ = 64, V_ = 64;
constexpr int CHUNK = 16;
constexpr int NCHUNK = T_ / CHUNK;
constexpr int THREADS = 128;   // 4 waves of 32
constexpr int CFLOATS = CHUNK * K_;   // 1024 floats per tensor per chunk

#if ASYNC_LDS
// Builtin prototype: (v4i addrspace(1)*, v4i addrspace(3)*, imm int offset, imm int cpol)
typedef int v4i __attribute__((vector_size(4 * sizeof(int))));
typedef __attribute__((address_space(1))) v4i gv4i;
typedef __attribute__((address_space(3))) v4i lv4i;

__device__ __forceinline__ lv4i* to_lds(const void* p) {
    // Flat pointers into the LDS aperture carry the LDS byte offset in addr[31:0].
    return (lv4i*)(uintptr_t)(uint32_t)(uintptr_t)p;
}
__device__ __forceinline__ gv4i* to_glob(const void* p) {
    return (gv4i*)(uintptr_t)p;
}
#endif

// Stage one 16x64 f32 chunk of each tensor into LDS (8 b128 transfers per wave).
__device__ __forceinline__ void stage_chunk(const float* __restrict__ gq,
                                            const float* __restrict__ gk,
                                            const float* __restrict__ gv,
                                            const float* __restrict__ gw,
                                            size_t cb,
                                            float* dq, float* dk, float* dv, float* dw,
                                            int tid)
{
#if ASYNC_LDS
    #pragma unroll
    for (int i = 0; i < 2; ++i) {
        const int f = (tid + i * THREADS) * 4;   // float index, 16B-aligned
        __builtin_amdgcn_global_load_async_to_lds_b128(to_glob(gq + cb + f), to_lds(dq + f), 0, 0);
        __builtin_amdgcn_global_load_async_to_lds_b128(to_glob(gk + cb + f), to_lds(dk + f), 0, 0);
        __builtin_amdgcn_global_load_async_to_lds_b128(to_glob(gv + cb + f), to_lds(dv + f), 0, 0);
        __builtin_amdgcn_global_load_async_to_lds_b128(to_glob(gw + cb + f), to_lds(dw + f), 0, 0);
    }
#else
    #pragma unroll
    for (int i = 0; i < 2; ++i) {
        const int idx = tid + i * THREADS;       // 0..255 float4
        ((float4*)dq)[idx] = ((const float4*)(gq + cb))[idx];
        ((float4*)dk)[idx] = ((const float4*)(gk + cb))[idx];
        ((float4*)dv)[idx] = ((const float4*)(gv + cb))[idx];
        ((float4*)dw)[idx] = ((const float4*)(gw + cb))[idx];
    }
#endif
}

template <int N>
__device__ __forceinline__ void wait_async()
{
#if ASYNC_LDS
    __builtin_amdgcn_s_wait_asynccnt((unsigned short)N);
#endif
}

// One workgroup per (b,h). Wave j owns v-columns [16j, 16j+16) of the 64x64 state.
__global__ __launch_bounds__(THREADS, 1)
void rwkv6_chunked_wmma(const float* __restrict__ gr, const float* __restrict__ gk,
                        const float* __restrict__ gv, const float* __restrict__ gw,
                        const float* __restrict__ gu, const float* __restrict__ gh0,
                        float* __restrict__ go, float* __restrict__ ghf)
{
    __shared__ float sH [K_ * V_];        // state, k-major: sH[k*64 + v]
    __shared__ float sQ [2][CFLOATS];     // raw q -> q~  (q * Aex)        (ping-pong)
    __shared__ float sK [2][CFLOATS];     // raw k -> k^  (k / (Aex*decay))
    __shared__ float sV [2][CFLOATS];     // raw v
    __shared__ float sW [2][CFLOATS];     // raw w -> q*u*Aex*decay (diag helper)
    __shared__ float sKc[CFLOATS];        // k^ * Aend
    __shared__ float sS [CHUNK * CHUNK];  // masked intra-chunk attention
    __shared__ float sAend[K_];

    const int tid  = threadIdx.x;
    const int lane = tid & 31;
    const int wave = tid >> 5;            // 0..3  -> v-tile
    const int l16  = lane & 15;
    const int lh   = lane >> 4;           // 0/1 (lane half)
    const int vcol = wave * 16 + l16;     // this lane's v column (B/C operand N)

    const int bh = blockIdx.x;
    const int h  = bh % H_;
    const size_t base = (size_t)bh * T_ * K_;

    // initial state (flat 64x64, same layout as (K,V))
    for (int i = tid; i < K_ * V_; i += THREADS)
        sH[i] = gh0[(size_t)bh * K_ * V_ + i];

    const float u_k = (tid < K_) ? gu[h * K_ + tid] : 0.0f;

    // prologue: stage chunk 0 into buffer 0
    stage_chunk(gr, gk, gv, gw, base, sQ[0], sK[0], sV[0], sW[0], tid);

    for (int ch = 0; ch < NCHUNK; ++ch) {
        const int cur = ch & 1;
        const int nxt = cur ^ 1;
        const size_t cb = base + (size_t)ch * CFLOATS;

        // stage next chunk into the other buffer, then wait for current chunk
        if (ch + 1 < NCHUNK) {
            stage_chunk(gr, gk, gv, gw, cb + CFLOATS,
                        sQ[nxt], sK[nxt], sV[nxt], sW[nxt], tid);
            wait_async<8>();   // 8 just issued outstanding; current chunk done
        } else {
            wait_async<0>();
        }
        __syncthreads();       // publish this wave's staged data to all waves

        float* bQ = sQ[cur];
        float* bK = sK[cur];
        float* bV = sV[cur];
        float* bW = sW[cur];

        // ---- per-k cumulative decays + operand transforms (threads 0..63) ----
        if (tid < K_) {
            float run = 1.0f;                    // inclusive cumprod of decay
            float khbuf[CHUNK];
            #pragma unroll
            for (int t = 0; t < CHUNK; ++t) {
                const int idx = t * K_ + tid;
                const float d   = __expf(bW[idx]);   // decay_t[k], in (e^-2, 1)
                const float Aex = run;               // prod_{s<t} decay
                const float qv  = bQ[idx];
                const float kv  = bK[idx];
                const float Ain = Aex * d;           // prod_{s<=t}
                const float kh  = kv / Ain;          // k^
                bW[idx] = qv * u_k * Ain;            // q*u*A_{t+1}: diag(qW.k^T)=q*u*k
                bQ[idx] = qv * Aex;                  // q~
                bK[idx] = kh;
                khbuf[t] = kh;
                run = Ain;
            }
            sAend[tid] = run;
            #pragma unroll
            for (int t = 0; t < CHUNK; ++t)
                sKc[t * K_ + tid] = khbuf[t] * run;  // decay from s+1 to chunk end
        }
        __syncthreads();

        // ---- intra-chunk attention S (wave 0 only; EXEC uniform in-wave) ----
        if (wave == 0) {
            v8f S  = {};
            v8f Su = {};
            #pragma unroll
            for (int c = 0; c < 16; ++c) {
                const int kk = 4 * c + 2 * lh;       // reduction index (A/B layout)
                v2f a, au, b;
                a[0]  = bQ[l16 * K_ + kk];  a[1]  = bQ[l16 * K_ + kk + 1];
                au[0] = bW[l16 * K_ + kk];  au[1] = bW[l16 * K_ + kk + 1];
                b[0]  = bK[l16 * K_ + kk];  b[1]  = bK[l16 * K_ + kk + 1];
                S  = WMMA_F32(a,  b, S);             // q~ . k^T
                Su = WMMA_F32(au, b, Su);            // diag source
            }
            #pragma unroll
            for (int rr = 0; rr < 8; ++rr) {         // C layout: row m, col s
                const int m = rr + 8 * lh;
                const int s = l16;
                const float val = (s < m) ? S[rr] : ((s == m) ? Su[rr] : 0.0f);
                sS[m * 16 + s] = val;
            }
        }
        __syncthreads();

        // ---- output: o = q~ . H0 + S . V   (wave j -> v-tile j) ----
        {
            v8f oacc = {};
            #pragma unroll
            for (int c = 0; c < 16; ++c) {           // cross-chunk: K = 64
                const int kk = 4 * c + 2 * lh;
                v2f a, b;
                a[0] = bQ[l16 * K_ + kk];      a[1] = bQ[l16 * K_ + kk + 1];
                b[0] = sH[kk * V_ + vcol];     b[1] = sH[(kk + 1) * V_ + vcol];
                oacc = WMMA_F32(a, b, oacc);
            }
            #pragma unroll
            for (int c = 0; c < 4; ++c) {            // intra-chunk: K = 16
                const int ss = 4 * c + 2 * lh;
                v2f a, b;
                a[0] = sS[l16 * 16 + ss];      a[1] = sS[l16 * 16 + ss + 1];
                b[0] = bV[ss * V_ + vcol];     b[1] = bV[(ss + 1) * V_ + vcol];
                oacc = WMMA_F32(a, b, oacc);
            }
            #pragma unroll
            for (int rr = 0; rr < 8; ++rr) {         // C layout rows -> timesteps
                const int t = rr + 8 * lh;
                go[cb + (size_t)t * K_ + vcol] = oacc[rr];
            }
        }

        // ---- state update: H = Aend (.) H0 + Kc^T . V  (columns wave-private) ----
        #pragma unroll
        for (int i = 0; i < 4; ++i) {                // k-tile
            v8f hacc;
            #pragma unroll
            for (int rr = 0; rr < 8; ++rr) {
                const int kk = 16 * i + rr + 8 * lh;
                hacc[rr] = sH[kk * V_ + vcol] * sAend[kk];
            }
            #pragma unroll
            for (int c = 0; c < 4; ++c) {            // reduction over s: K = 16
                const int ss = 4 * c + 2 * lh;
                v2f a, b;
                a[0] = sKc[ss * K_ + 16 * i + l16];
                a[1] = sKc[(ss + 1) * K_ + 16 * i + l16];
                b[0] = bV[ss * V_ + vcol];
                b[1] = bV[(ss + 1) * V_ + vcol];
                hacc = WMMA_F32(a, b, hacc);
            }
            #pragma unroll
            for (int rr = 0; rr < 8; ++rr) {
                const int kk = 16 * i + rr + 8 * lh;
                sH[kk * V_ + vcol] = hacc[rr];
            }
        }

        // all consumers of buf[cur] done before it is re-staged next iteration
        __syncthreads();
    }

    // final state out
    for (int i = tid; i < K_ * V_; i += THREADS)
        ghf[(size_t)bh * K_ * V_ + i] = sH[i];
}

extern "C" void kernel_launch(void* const* d_in, const int* in_sizes, int n_in,
                              void* d_out, int out_size, void* d_ws, size_t ws_size,
                              hipStream_t stream) {
    const float* r  = (const float*)d_in[0];
    const float* k  = (const float*)d_in[1];
    const float* v  = (const float*)d_in[2];
    const float* w  = (const float*)d_in[3];
    const float* u  = (const float*)d_in[4];
    const float* h0 = (const float*)d_in[5];
    float* o  = (float*)d_out;
    float* hf = o + (size_t)B_ * H_ * T_ * V_;   // tuple output: (o, final_state)

    dim3 grid(B_ * H_);       // 128 workgroups, one per (b,h)
    dim3 block(THREADS);      // 4 waves
    hipLaunchKernelGGL(rwkv6_chunked_wmma, grid, block, 0, stream,
                       r, k, v, w, u, h0, o, hf);
}